// RGCNModel_1846835938035
// MI455X (gfx1250) — compile-verified
//
#include <hip/hip_runtime.h>
#include <hip/hip_bf16.h>

// CDNA5 / gfx1250: wave32, WMMA fp32 path.
typedef __attribute__((ext_vector_type(2))) float v2f;
typedef __attribute__((ext_vector_type(8))) float v8f;

// ---------------------------------------------------------------------------
// Dense transform: for mat in [0,8): T[mat] = act(X) @ Wrel[mat]
//                  for mat == 8    : Hroot  = act(X) @ Wroot + bias
// act(X) = X when mask==nullptr, else relu(X)*mask (fused dropout for layer 2).
// One workgroup = 8 waves computes a TILE_M x COLS slab.
//   COLS=128 -> 1 row sub-tile  x 8 col tiles (TILE_M=16)
//   COLS=64  -> 2 row sub-tiles x 4 col tiles (TILE_M=32)
// X tile staged in LDS k-major with +1 padding (bank-conflict-free A reads,
// lowers to paired ds_load_2addr_b32).
// ---------------------------------------------------------------------------
template <int COLS>
__global__ __launch_bounds__(256) void rgcn_gemm_kernel(
    const float* __restrict__ X,     // [N,128]
    const float* __restrict__ mask,  // [N,128] or nullptr
    const float* __restrict__ Wrel,  // [8,128,COLS]
    const float* __restrict__ Wroot, // [128,COLS]
    const float* __restrict__ bias,  // [COLS]
    float* __restrict__ Trel,        // [8,N,COLS]
    float* __restrict__ Hroot,       // [N,COLS]
    int N)
{
    constexpr int CT     = COLS / 16;   // col tiles per block (8 or 4)
    constexpr int RT     = 8 / CT;      // row sub-tiles (1 or 2)
    constexpr int TILE_M = RT * 16;     // 16 or 32
    constexpr int LDA    = TILE_M + 1;  // pad to kill LDS bank conflicts
    __shared__ float xs[128 * LDA];

    const int  tid     = threadIdx.x;
    const int  rowBase = blockIdx.x * TILE_M;
    const int  mat     = blockIdx.y;    // 0..7 = relations, 8 = root
    const bool full    = (rowBase + TILE_M) <= N;  // workgroup-uniform

    // Stage X tile into LDS, transposed to k-major. Coalesced global reads.
    if (full) {
        #pragma unroll
        for (int it = 0; it < TILE_M * 128 / 256; ++it) {
            int idx = tid + it * 256;
            int m   = idx >> 7;     // / 128
            int k   = idx & 127;
            size_t g = (size_t)(rowBase + m) * 128 + k;
            float  v = X[g];
            if (mask) v = fmaxf(v, 0.0f) * mask[g];   // fused relu*dropout
            xs[k * LDA + m] = v;
        }
    } else {
        #pragma unroll
        for (int it = 0; it < TILE_M * 128 / 256; ++it) {
            int idx = tid + it * 256;
            int m   = idx >> 7;
            int k   = idx & 127;
            int row = rowBase + m;
            float v = 0.0f;
            if (row < N) {
                size_t g = (size_t)row * 128 + k;
                v = X[g];
                if (mask) v = fmaxf(v, 0.0f) * mask[g];
            }
            xs[k * LDA + m] = v;
        }
    }
    __syncthreads();

    const int wave   = tid >> 5;
    const int lane   = tid & 31;
    const int half   = lane >> 4;       // 0: K=k0..k0+1, 1: K=k0+2..k0+3
    const int mm     = lane & 15;
    const int rowSub = (wave / CT) * 16;
    const int n0     = (wave % CT) * 16;

    const float* __restrict__ Wp =
        (mat < 8) ? (Wrel + (size_t)mat * 128 * COLS) : Wroot;

    v8f acc = {};
    #pragma unroll 4
    for (int k0 = 0; k0 < 128; k0 += 4) {
        const int ka = k0 + 2 * half;
        v2f a, b;
        // A fragment (16x4 fp32, ISA layout): lane(mm) = M, half selects K pair
        a.x = xs[ka * LDA + rowSub + mm];
        a.y = xs[(ka + 1) * LDA + rowSub + mm];
        // B fragment (4x16 fp32): lane(mm) = N column, half selects K pair
        b.x = Wp[(size_t)ka * COLS + n0 + mm];
        b.y = Wp[(size_t)(ka + 1) * COLS + n0 + mm];
        acc = __builtin_amdgcn_wmma_f32_16x16x4_f32(
            /*neg_a=*/false, a, /*neg_b=*/false, b,
            /*c_mod=*/(short)0, acc, /*reuse_a=*/false, /*reuse_b=*/false);
    }

    const float bval = (mat == 8) ? bias[n0 + mm] : 0.0f;
    float* __restrict__ outp =
        (mat < 8) ? (Trel + (size_t)mat * N * COLS) : Hroot;

    // C/D layout: VGPR v, lane half -> M = v + 8*half, N = mm.
    // Single base pointer + immediate offsets -> clean clause of stores.
    const int row0 = rowBase + rowSub + 8 * half;
    float* __restrict__ op = outp + (size_t)row0 * COLS + n0 + mm;
    if (full) {
        #pragma unroll
        for (int v = 0; v < 8; ++v) op[(size_t)v * COLS] = acc[v] + bval;
    } else {
        #pragma unroll
        for (int v = 0; v < 8; ++v)
            if (row0 + v < N) op[(size_t)v * COLS] = acc[v] + bval;
    }
}

// ---------------------------------------------------------------------------
// Per-(relation,target) in-degree, then inv-clip in place.
// ---------------------------------------------------------------------------
__global__ __launch_bounds__(256) void rgcn_zero_kernel(float* p, int n)
{
    int i = blockIdx.x * blockDim.x + threadIdx.x;
    if (i < n) p[i] = 0.0f;
}

__global__ __launch_bounds__(256) void rgcn_deg_count_kernel(
    const int* __restrict__ edst, const int* __restrict__ etype,
    float* __restrict__ deg, int N, int E)
{
    int e = blockIdx.x * blockDim.x + threadIdx.x;
    if (e >= E) return;
    atomicAdd(&deg[(size_t)etype[e] * N + edst[e]], 1.0f);
}

__global__ __launch_bounds__(256) void rgcn_deg_finalize_kernel(
    float* __restrict__ deg, int n)
{
    int i = blockIdx.x * blockDim.x + threadIdx.x;
    if (i < n) deg[i] = 1.0f / fmaxf(deg[i], 1.0f);
}

// ---------------------------------------------------------------------------
// Edge gather/scatter: one wave per edge. COLS/32 floats per lane,
// vector gather from the transformed table, atomic fp32 scatter into target.
// ---------------------------------------------------------------------------
template <int COLS>
__global__ __launch_bounds__(256) void rgcn_msg_kernel(
    const float* __restrict__ T,      // [8,N,COLS]
    const int* __restrict__ esrc, const int* __restrict__ edst,
    const int* __restrict__ etype,
    const float* __restrict__ invdeg, // [8,N]
    float* __restrict__ out,          // [N,COLS] (accumulated)
    int N, int E)
{
    constexpr int PER = COLS / 32;
    int gid  = blockIdx.x * blockDim.x + threadIdx.x;
    int e    = gid >> 5;
    int lane = threadIdx.x & 31;
    if (e >= E) return;

    const int   rel = etype[e];
    const int   s   = esrc[e];
    const int   d   = edst[e];
    const float c   = invdeg[(size_t)rel * N + d];

    const float* __restrict__ sp = T + ((size_t)rel * N + s) * COLS + lane * PER;
    float* __restrict__ dp       = out + (size_t)d * COLS + lane * PER;

    float vals[PER];
    if constexpr (PER == 4) {
        float4 v = *reinterpret_cast<const float4*>(sp);
        vals[0] = v.x; vals[1] = v.y; vals[2] = v.z; vals[3] = v.w;
    } else {
        float2 v = *reinterpret_cast<const float2*>(sp);
        vals[0] = v.x; vals[1] = v.y;
    }
    #pragma unroll
    for (int i = 0; i < PER; ++i) atomicAdd(&dp[i], vals[i] * c);
}

// ---------------------------------------------------------------------------
extern "C" void kernel_launch(void* const* d_in, const int* in_sizes, int n_in,
                              void* d_out, int out_size, void* d_ws, size_t ws_size,
                              hipStream_t stream)
{
    const float* x     = (const float*)d_in[0]; // [N,128]
    const int*   eidx  = (const int*)d_in[1];   // [2,E]
    const int*   etype = (const int*)d_in[2];   // [E]
    const float* W1    = (const float*)d_in[3]; // [8,128,128]
    const float* root1 = (const float*)d_in[4]; // [128,128]
    const float* b1    = (const float*)d_in[5]; // [128]
    const float* W2    = (const float*)d_in[6]; // [8,128,64]
    const float* root2 = (const float*)d_in[7]; // [128,64]
    const float* b2    = (const float*)d_in[8]; // [64]
    const float* dmask = (const float*)d_in[9]; // [N,128]

    const int N = in_sizes[0] / 128; // 50000
    const int E = in_sizes[2];       // 400000
    const int R = 8;

    const int* esrc = eidx;       // edge_index row 0
    const int* edst = eidx + E;   // edge_index row 1

    // Workspace layout (fp32):
    //   Tbuf: R*N*128   (layer-1 transformed; reused for layer-2 [R,N,64])
    //   h   : N*128     (layer-1 output, pre-activation; activation fused
    //                    into layer-2 staging)
    //   deg : R*N       (degree counts -> inv coefficients, shared by layers)
    float* Tbuf = (float*)d_ws;
    float* h    = Tbuf + (size_t)R * N * 128;
    float* deg  = h + (size_t)N * 128;

    // --- degree coefficients (relation x target), shared by both layers ---
    {
        int n = R * N;
        rgcn_zero_kernel<<<(n + 255) / 256, 256, 0, stream>>>(deg, n);
        rgcn_deg_count_kernel<<<(E + 255) / 256, 256, 0, stream>>>(edst, etype, deg, N, E);
        rgcn_deg_finalize_kernel<<<(n + 255) / 256, 256, 0, stream>>>(deg, n);
    }

    // --- layer 1: T1[r] = x @ W1[r];  h = x @ root1 + b1 ---
    {
        dim3 grid((N + 15) / 16, 9);
        rgcn_gemm_kernel<128><<<grid, 256, 0, stream>>>(
            x, nullptr, W1, root1, b1, Tbuf, h, N);
    }
    // --- layer 1: mean-aggregated messages into h ---
    {
        long long threads = (long long)E * 32;
        rgcn_msg_kernel<128><<<(int)((threads + 255) / 256), 256, 0, stream>>>(
            Tbuf, esrc, edst, etype, deg, h, N, E);
    }
    // --- layer 2 (relu*dropout fused into staging):
    //     T2[r] = act(h) @ W2[r];  out = act(h) @ root2 + b2 ---
    {
        dim3 grid((N + 31) / 32, 9);
        rgcn_gemm_kernel<64><<<grid, 256, 0, stream>>>(
            h, dmask, W2, root2, b2, Tbuf, (float*)d_out, N);
    }
    // --- layer 2: messages into out ---
    {
        long long threads = (long long)E * 32;
        rgcn_msg_kernel<64><<<(int)((threads + 255) / 256), 256, 0, stream>>>(
            Tbuf, esrc, edst, etype, deg, (float*)d_out, N, E);
    }
}